// BaseRouter_22488448761978
// MI455X (gfx1250) — compile-verified
//
#include <hip/hip_runtime.h>
#include <stdint.h>

// Problem constants (reference: B=8, T=4096, D=2048, k = int(4096*0.125) = 512)
#define B_      8
#define T_      4096
#define D_      2048
#define K_      512

// d_out layout (floats), concatenated in reference return order:
// output[B*T*D], selected_states[B*K*D], batch_idx[B*K], token_idx[B*K], selected_scores[B*K]
#define SEL_OFF  ((long long)B_ * T_ * D_)                 // 67,108,864
#define BI_OFF   (SEL_OFF + (long long)B_ * K_ * D_)       // 75,497,472
#define TI_OFF   (BI_OFF + (long long)B_ * K_)             // 75,501,568
#define SC_OFF   (TI_OFF + (long long)B_ * K_)             // 75,505,664

// Fused-kernel block partition (single launch; blocks dispatched in order):
//   [0, B_)                         top-k producers (resident first)
//   [B_, B_+COPY_BLOCKS)            256MB copy via async-LDS DMA path
//   [B_+COPY_BLOCKS, ...)           gather consumers (dispatched last, spin on flags)
#define COPY_BLOCKS 4096   // 4096 blocks * 1024 thr * 4 float4 = 16,777,216 float4 = B*T*D floats
#define GATH_BLOCKS 2048   // 2 selected rows per block (1024 thr = 2 * 512 float4)
#define MAGIC 0x13579BDF

__global__ __launch_bounds__(1024) void fused_router_kernel(const float* __restrict__ scores,
                                                            const float* __restrict__ hidden,
                                                            float* __restrict__ out,
                                                            int* __restrict__ ws) {
    __shared__ unsigned long long keys[T_];   // 32 KB: bitonic sort keys (top-k blocks)
    __shared__ float4 stage[4096];            // 64 KB: async DMA staging (copy blocks)

    int* __restrict__ done = ws + (B_ * K_);  // 8 publish words after the row-index table
    const int      tid = threadIdx.x;
    const unsigned bx  = blockIdx.x;

    if (bx < B_) {
        // ------------------------------------------------------------------
        // Role 1: exact top-512 of one score row via bitonic sort in LDS.
        // key = (order-preserving u32 of f32) << 32 | ~index  => descending
        // sort reproduces jax.lax.top_k order incl. lower-index tie break.
        // ------------------------------------------------------------------
        const int b = (int)bx;
        for (int i = tid; i < T_; i += 1024) {
            float v = scores[(long long)b * T_ + i];
            unsigned u = __float_as_uint(v);
            u = (u & 0x80000000u) ? ~u : (u | 0x80000000u);
            keys[i] = ((unsigned long long)u << 32) | (unsigned)(~(unsigned)i);
        }
        __syncthreads();

        for (int k = 2; k <= T_; k <<= 1) {
            for (int j = k >> 1; j > 0; j >>= 1) {
                for (int i = tid; i < T_; i += 1024) {
                    int l = i ^ j;
                    if (l > i) {
                        bool desc = ((i & k) == 0);
                        unsigned long long a = keys[i];
                        unsigned long long c = keys[l];
                        bool sw = desc ? (a < c) : (a > c);
                        if (sw) { keys[i] = c; keys[l] = a; }
                    }
                }
                __syncthreads();
            }
        }

        if (tid < K_) {
            unsigned long long key = keys[tid];
            unsigned u  = (unsigned)(key >> 32);
            unsigned fu = (u & 0x80000000u) ? (u ^ 0x80000000u) : ~u;
            float val = __uint_as_float(fu);
            int   idx = (int)(~(unsigned)key);
            int   o   = b * K_ + tid;
            out[SC_OFF + o] = val;
            out[TI_OFF + o] = (float)idx;
            out[BI_OFF + o] = (float)b;
            // row table consumed by gather blocks: publish at agent scope
            __hip_atomic_store(&ws[o], b * T_ + idx,
                               __ATOMIC_RELAXED, __HIP_MEMORY_SCOPE_AGENT);
        }
        __syncthreads();
        if (tid == 0) {
            __hip_atomic_store(&done[b], MAGIC,
                               __ATOMIC_RELEASE, __HIP_MEMORY_SCOPE_AGENT);
        }
    } else if (bx < B_ + COPY_BLOCKS) {
        // ------------------------------------------------------------------
        // Role 2: output = hidden_states (256 MB) through the CDNA5 async
        // DMA path: global_load_async_to_lds_b128 -> s_wait_asynccnt ->
        // global_store_async_from_lds_b128. 4 x 16B outstanding per lane.
        // ------------------------------------------------------------------
        const unsigned  cb   = bx - B_;
        const long long base = (long long)cb * 4096;   // float4 units
        const float4* __restrict__ src = (const float4*)hidden;
        float4* __restrict__ dst = (float4*)out;
        // Low 32 bits of the flat address of an LDS object == LDS byte offset.
        const unsigned lds_base = (unsigned)(uintptr_t)(&stage[0]);

#pragma unroll
        for (int c = 0; c < 4; ++c) {
            unsigned slot = (unsigned)c * 1024u + (unsigned)tid;
            unsigned lds_addr = lds_base + slot * 16u;
            const float4* g = src + base + slot;
            asm volatile("global_load_async_to_lds_b128 %0, %1, off"
                         :: "v"(lds_addr), "v"(g) : "memory");
        }
        // async-load LDS writes vs async-store LDS reads are unordered: wait.
        asm volatile("s_wait_asynccnt 0" ::: "memory");
#pragma unroll
        for (int c = 0; c < 4; ++c) {
            unsigned slot = (unsigned)c * 1024u + (unsigned)tid;
            unsigned lds_addr = lds_base + slot * 16u;
            float4* g = dst + base + slot;
            asm volatile("global_store_async_from_lds_b128 %0, %1, off"
                         :: "v"(g), "v"(lds_addr) : "memory");
        }
        asm volatile("s_wait_asynccnt 0" ::: "memory");
    } else {
        // ------------------------------------------------------------------
        // Role 3: gather selected rows (2 rows x 2048 floats per block).
        // Wait for all 8 top-k producers first (they were dispatched first
        // and depend on nobody, so the spin cannot deadlock).
        // ------------------------------------------------------------------
        const unsigned gb = bx - (B_ + COPY_BLOCKS);
        if (tid == 0) {
            for (int i = 0; i < B_; ++i) {
                while (__hip_atomic_load(&done[i], __ATOMIC_ACQUIRE,
                                         __HIP_MEMORY_SCOPE_AGENT) != MAGIC) {
                    __builtin_amdgcn_s_sleep(1);
                }
            }
        }
        __syncthreads();

        const int r = (int)(gb * 2) + (tid >> 9);   // selected-row id, 0..B*K-1
        const int e = tid & 511;                    // float4 lane within row
        const long long srow =
            (long long)__hip_atomic_load(&ws[r], __ATOMIC_ACQUIRE,
                                         __HIP_MEMORY_SCOPE_AGENT);
        const float4* __restrict__ src = (const float4*)hidden + srow * (D_ / 4);
        float4* __restrict__ dst = (float4*)(out + SEL_OFF) + (long long)r * (D_ / 4);
        dst[e] = src[e];
    }
}

extern "C" void kernel_launch(void* const* d_in, const int* in_sizes, int n_in,
                              void* d_out, int out_size, void* d_ws, size_t ws_size,
                              hipStream_t stream) {
    (void)in_sizes; (void)n_in; (void)out_size; (void)ws_size;
    const float* scores = (const float*)d_in[0];   // [B, T] f32
    const float* hidden = (const float*)d_in[1];   // [B, T, D] f32
    float* out   = (float*)d_out;
    int*   wsidx = (int*)d_ws;                     // B*K row indices + B_ flag words

    const int grid = B_ + COPY_BLOCKS + GATH_BLOCKS;   // 6152 blocks
    fused_router_kernel<<<grid, 1024, 0, stream>>>(scores, hidden, out, wsidx);
}